// SparseProjector_84069689852430
// MI455X (gfx1250) — compile-verified
//
#include <hip/hip_runtime.h>
#include <cstdint>
#include <cstddef>

// Problem constants (from reference)
#define B_   2
#define SRC_ 262144
#define DST_ 262144
#define E_   1048576
#define F_   128
#define EPSF 1e-8f

#define WAVES_PER_BLOCK 8   // 256 threads, wave32

// ---------------------------------------------------------------------------
// HOT KERNEL FIRST so the disasm snippet shows it.
//
// Wave-per-edge scatter with a 2-deep async (global -> LDS) pipeline.
// Lane l owns features [4l, 4l+4). One GLOBAL_LOAD_ASYNC_TO_LDS_B128 moves a
// full 512-byte x-row (32 lanes x 16B) per batch; ASYNCcnt tracks completion.
// ---------------------------------------------------------------------------
__global__ __launch_bounds__(256)
void scatter_kernel(const float* __restrict__ x,
                    const float* __restrict__ w,
                    const int* __restrict__ srcI,
                    const int* __restrict__ dstI,
                    const float* __restrict__ norm,
                    float* __restrict__ out,
                    int nwaves_total) {
  __shared__ __align__(16) float lbuf[WAVES_PER_BLOCK][2][B_][F_];

  const int lane  = threadIdx.x & 31;
  const int wv    = threadIdx.x >> 5;
  const int gwave = blockIdx.x * WAVES_PER_BLOCK + wv;

  // Stage both batches' x-rows of edge e into LDS slot (async DMA, no VGPR data).
  auto stage = [&](int e, int slot) {
    int s = srcI[e];
#pragma unroll
    for (int b = 0; b < B_; ++b) {
      const float* gsrc = x + ((size_t)b * SRC_ + (size_t)s) * F_ + (size_t)lane * 4;
      uint64_t ga = (uint64_t)(uintptr_t)gsrc;
      uint32_t la = (uint32_t)(uintptr_t)(&lbuf[wv][slot][b][lane * 4]);
      asm volatile("global_load_async_to_lds_b128 %0, %1, off"
                   :: "v"(la), "v"(ga) : "memory");
    }
  };

  int e = gwave;
  if (e < E_) stage(e, 0);

  int slot = 0;
  for (; e < E_; e += nwaves_total, slot ^= 1) {
    const int en = e + nwaves_total;
    const bool has_next = (en < E_);
    if (has_next) stage(en, slot ^ 1);          // overlap next gather with this edge

    const int   d  = dstI[e];
    const float wn = w[e] / (norm[d] + EPSF);

    // Current edge issued 2 async instructions; next edge (if staged) adds 2 more.
    // Async loads complete in order, so waiting down to the newer pair suffices.
    if (has_next) asm volatile("s_wait_asynccnt 0x2" ::: "memory");
    else          asm volatile("s_wait_asynccnt 0x0" ::: "memory");

#pragma unroll
    for (int b = 0; b < B_; ++b) {
      const float4 v = *reinterpret_cast<const float4*>(&lbuf[wv][slot][b][lane * 4]);
      float* orow = out + ((size_t)b * DST_ + (size_t)d) * F_ + (size_t)lane * 4;
      (void)__hip_atomic_fetch_add(&orow[0], v.x * wn, __ATOMIC_RELAXED, __HIP_MEMORY_SCOPE_AGENT);
      (void)__hip_atomic_fetch_add(&orow[1], v.y * wn, __ATOMIC_RELAXED, __HIP_MEMORY_SCOPE_AGENT);
      (void)__hip_atomic_fetch_add(&orow[2], v.z * wn, __ATOMIC_RELAXED, __HIP_MEMORY_SCOPE_AGENT);
      (void)__hip_atomic_fetch_add(&orow[3], v.w * wn, __ATOMIC_RELAXED, __HIP_MEMORY_SCOPE_AGENT);
    }
  }
}

// ---------------------------------------------------------------------------
// norm[d] = sum of weights of edges whose dst == d  (agent-scope f32 atomics)
// ---------------------------------------------------------------------------
__global__ void norm_kernel(const float* __restrict__ w,
                            const int* __restrict__ dstI,
                            float* __restrict__ norm) {
  int e = blockIdx.x * blockDim.x + threadIdx.x;
  if (e < E_) {
    (void)__hip_atomic_fetch_add(&norm[dstI[e]], w[e],
                                 __ATOMIC_RELAXED, __HIP_MEMORY_SCOPE_AGENT);
  }
}

// ---------------------------------------------------------------------------
// Zero a buffer with 16-byte stores (used for norm scratch and d_out).
// ---------------------------------------------------------------------------
__global__ void zero_kernel(float4* __restrict__ p, int n4) {
  int i = blockIdx.x * blockDim.x + threadIdx.x;
  int stride = gridDim.x * blockDim.x;
  float4 z = make_float4(0.f, 0.f, 0.f, 0.f);
  for (; i < n4; i += stride) p[i] = z;
}

// ---------------------------------------------------------------------------
extern "C" void kernel_launch(void* const* d_in, const int* in_sizes, int n_in,
                              void* d_out, int out_size, void* d_ws, size_t ws_size,
                              hipStream_t stream) {
  (void)in_sizes; (void)n_in; (void)out_size; (void)ws_size;

  const float* x    = (const float*)d_in[0];          // (B, SRC, F) f32
  const float* wgt  = (const float*)d_in[1];          // (E,) f32
  const int*   ei   = (const int*)d_in[2];            // (2, E) i32
  const int*   srcI = ei;
  const int*   dstI = ei + E_;
  float*       out  = (float*)d_out;                  // (B, DST, F) f32
  float*       norm = (float*)d_ws;                   // DST f32 scratch

  // 1) zero per-destination weight sums (d_ws is not re-poisoned between calls)
  zero_kernel<<<256, 256, 0, stream>>>((float4*)norm, DST_ / 4);
  // 2) zero output (scatter accumulates with atomics)
  zero_kernel<<<4096, 256, 0, stream>>>((float4*)out, (B_ * DST_ * F_) / 4);
  // 3) norm[d] = segment_sum(weights, dst)
  norm_kernel<<<E_ / 256, 256, 0, stream>>>(wgt, dstI, norm);
  // 4) normalized scatter-add of x rows
  const int blocks = 4096;
  scatter_kernel<<<blocks, 256, 0, stream>>>(x, wgt, srcI, dstI, norm, out,
                                             blocks * WAVES_PER_BLOCK);
}